// VolumeRenderer_10763188044530
// MI455X (gfx1250) — compile-verified
//
#include <hip/hip_runtime.h>
#include <stdint.h>

#define N_RAYS 65536
#define NS 192
#define RPB 8          // rays per block (8 waves of 32)
#define SPL 6          // samples per lane: 192 / 32
#define BLOCK 256

typedef unsigned int u32x4 __attribute__((ext_vector_type(4)));
typedef int          i32x8 __attribute__((ext_vector_type(8)));
typedef int          i32x4 __attribute__((ext_vector_type(4)));

#if __has_builtin(__builtin_amdgcn_tensor_load_to_lds)
#define HAVE_TDM 1
#else
#define HAVE_TDM 0
#endif

#if HAVE_TDM
// 1-D contiguous DWORD copy global -> LDS via Tensor Data Mover.
// D# packing per CDNA5 ISA ch.8: group0 = {count/type/lds_addr/global_addr},
// group1 = {data_size=4B, tensor_dim0, tensor_dim1=1, tile_dim0=n, stride=n}.
__device__ __forceinline__ void tdm_load_1d(uint32_t lds_off, const void* gptr,
                                            uint32_t ndwords) {
  uint64_t ga = (uint64_t)(uintptr_t)gptr;
  u32x4 g0;
  g0[0] = 1u;                                                  // count=1, user desc
  g0[1] = lds_off;                                             // lds_addr (bytes)
  g0[2] = (uint32_t)(ga & 0xFFFFFFFFu);                        // global_addr lo
  g0[3] = ((uint32_t)(ga >> 32) & 0x01FFFFFFu) | (2u << 30);   // ga hi25 | type=2
  i32x8 g1;
  g1[0] = (int)(2u << 16);                                     // data_size = 4 bytes
  g1[1] = (int)((ndwords & 0xFFFFu) << 16);                    // tensor_dim0[15:0]
  g1[2] = (int)(((ndwords >> 16) & 0xFFFFu) | (1u << 16));     // dim0 hi | tensor_dim1=1
  g1[3] = (int)((ndwords & 0xFFFFu) << 16);                    // tile_dim0 = ndwords
  g1[4] = 0;                                                   // tile_dim1=0 (1-D), tile_dim2=0
  g1[5] = (int)ndwords;                                        // tensor_dim0_stride lo
  g1[6] = 0;
  g1[7] = 0;
  i32x4 z4 = {0, 0, 0, 0};                                     // groups 2/3 unused (<=2D)
  i32x8 z8 = {0, 0, 0, 0, 0, 0, 0, 0};
  __builtin_amdgcn_tensor_load_to_lds(g0, g1, z4, z4, z8, 0);
}
#endif

__device__ __forceinline__ float fast_sigmoid(float x) {
  return 1.0f / (1.0f + __expf(-x));
}

__global__ __launch_bounds__(BLOCK) void VolumeRenderer_kernel(
    const float* __restrict__ depth, const float* __restrict__ rgb,
    const float* __restrict__ sigma, float* __restrict__ out) {
  __shared__ float s_depth[RPB * NS];          // 6144 B
  __shared__ float s_sigma[RPB * NS];          // 6144 B
  __shared__ float s_rgb[RPB * NS * 3];        // 18432 B

  const int tid  = threadIdx.x;
  const int lane = tid & 31;
  const int wave = tid >> 5;
  const int r0   = blockIdx.x * RPB;

#if HAVE_TDM
  // Spread the four DMA descriptors across four waves: TDM ops from the SAME
  // wave are processed in-order, so per-wave issue gives concurrent transfers.
  // Each issuing wave waits on its own TENSORcnt; the barrier joins everyone.
  if (wave == 0) {
    tdm_load_1d((uint32_t)(uintptr_t)(void*)s_depth,
                depth + (size_t)r0 * NS, RPB * NS);
    __builtin_amdgcn_s_wait_tensorcnt(0);
  } else if (wave == 1) {
    tdm_load_1d((uint32_t)(uintptr_t)(void*)s_sigma,
                sigma + (size_t)r0 * NS, RPB * NS);
    __builtin_amdgcn_s_wait_tensorcnt(0);
  } else if (wave == 2) {
    tdm_load_1d((uint32_t)(uintptr_t)(void*)s_rgb,
                rgb + (size_t)r0 * NS * 3, RPB * NS * 3 / 2);
    __builtin_amdgcn_s_wait_tensorcnt(0);
  } else if (wave == 3) {
    tdm_load_1d((uint32_t)(uintptr_t)(void*)(s_rgb + RPB * NS * 3 / 2),
                rgb + (size_t)r0 * NS * 3 + RPB * NS * 3 / 2, RPB * NS * 3 / 2);
    __builtin_amdgcn_s_wait_tensorcnt(0);
  }
#else
  for (int i = tid; i < RPB * NS; i += BLOCK) {
    s_depth[i] = depth[(size_t)r0 * NS + i];
    s_sigma[i] = sigma[(size_t)r0 * NS + i];
  }
  for (int i = tid; i < RPB * NS * 3; i += BLOCK)
    s_rgb[i] = rgb[(size_t)r0 * NS * 3 + i];
#endif
  __syncthreads();

  const int    ray = r0 + wave;
  const float* sd  = s_depth + wave * NS;
  const float* ss  = s_sigma + wave * NS;
  const float* sc  = s_rgb   + wave * NS * 3;

  const int s0 = lane * SPL;

  // Load my 6 depths + the next sample's depth (for deltas).
  float d[SPL + 1];
#pragma unroll
  for (int k = 0; k < SPL; ++k) d[k] = sd[s0 + k];
  d[SPL] = (lane == 31) ? 0.0f : sd[s0 + SPL];

  // alpha_k, and exclusive prefix product of survival within the lane.
  float alpha[SPL], pex[SPL];
  float p = 1.0f;
#pragma unroll
  for (int k = 0; k < SPL; ++k) {
    float delta = (s0 + k == NS - 1) ? 1e10f : (d[k + 1] - d[k]);
    float sg = ss[s0 + k];
    sg = sg > 0.0f ? sg : 0.0f;
    float e = __expf(-sg * delta);       // EXPSCALE = 1
    alpha[k] = 1.0f - e;
    float surv = e + 1e-10f;             // 1 - alpha + EPS
    pex[k] = p;
    p *= surv;
  }

  // Wave-exclusive product scan over per-lane survival products -> transmittance
  float scan = p;
#pragma unroll
  for (int off = 1; off < 32; off <<= 1) {
    float v = __shfl_up(scan, off, 32);
    if (lane >= off) scan *= v;
  }
  float T = __shfl_up(scan, 1, 32);
  if (lane == 0) T = 1.0f;

  // Per-sample weights + partial sums; store weights (8B-aligned pairs).
  float acc = 0.0f, dsum = 0.0f, c0 = 0.0f, c1 = 0.0f, c2 = 0.0f;
  float wv[SPL];
#pragma unroll
  for (int k = 0; k < SPL; ++k) {
    float w = alpha[k] * T * pex[k];
    wv[k] = w;
    acc  += w;
    dsum += w * d[k];
    const float* c = sc + (size_t)(s0 + k) * 3;
    c0 += w * fast_sigmoid(c[0]);
    c1 += w * fast_sigmoid(c[1]);
    c2 += w * fast_sigmoid(c[2]);
  }
  {
    float2* w2 = (float2*)(out + (size_t)5 * N_RAYS + (size_t)ray * NS + s0);
#pragma unroll
    for (int j = 0; j < SPL / 2; ++j) w2[j] = make_float2(wv[2 * j], wv[2 * j + 1]);
  }

  // Wave butterfly reductions (5 values x 5 steps).
#pragma unroll
  for (int m = 16; m >= 1; m >>= 1) {
    acc  += __shfl_xor(acc,  m, 32);
    dsum += __shfl_xor(dsum, m, 32);
    c0   += __shfl_xor(c0,   m, 32);
    c1   += __shfl_xor(c1,   m, 32);
    c2   += __shfl_xor(c2,   m, 32);
  }

  if (lane == 0) {
    out[(size_t)ray * 3 + 0] = c0;                 // color
    out[(size_t)ray * 3 + 1] = c1;
    out[(size_t)ray * 3 + 2] = c2;
    out[(size_t)3 * N_RAYS + ray] = dsum;          // depth_out
    out[(size_t)4 * N_RAYS + ray] = acc;           // acc_map
  }
}

extern "C" void kernel_launch(void* const* d_in, const int* in_sizes, int n_in,
                              void* d_out, int out_size, void* d_ws, size_t ws_size,
                              hipStream_t stream) {
  (void)in_sizes; (void)n_in; (void)d_ws; (void)ws_size; (void)out_size;
  const float* depth = (const float*)d_in[0];
  const float* rgb   = (const float*)d_in[1];
  const float* sigma = (const float*)d_in[2];
  float* out = (float*)d_out;
  dim3 grid(N_RAYS / RPB);   // 8192 blocks
  dim3 block(BLOCK);         // 8 waves -> 8 rays per block
  VolumeRenderer_kernel<<<grid, block, 0, stream>>>(depth, rgb, sigma, out);
}